// SWTInverse_88510686036772
// MI455X (gfx1250) — compile-verified
//
#include <hip/hip_runtime.h>

// ---------------------------------------------------------------------------
// Inverse SWT (db2, 2 levels) for N=8, C=1, H=W=1024 float32.
//
// Level math (derived from the reference):
//   out(r,c) = 0.5 * sum over (u,v) in [0,4)^2 with (u+v+t_r+t_c) even of
//       W_b[u,v] * Band_b[(r+s(u-2)) mod H, (c+s(v-2)) mod W]
//   W_A = LO[u]LO[v], W_H = HI[u]LO[v], W_V = LO[u]HI[v], W_D = HI[u]HI[v]
//   For s=2 only pixels with r%2==c%2 are computed; others pass ll through.
//
// Level-0 kernel uses the identity
//   sum_{parity} = 0.25*(S_full + (-1)^{r+c} S_alt)
// where S_full/S_alt are separable 4-tap convs (alt = sign-alternated taps),
// and performs the row-filter pass with V_WMMA_F32_16X16X4_F32 (exact f32).
// ---------------------------------------------------------------------------

typedef __attribute__((ext_vector_type(2))) float v2f;
typedef __attribute__((ext_vector_type(8))) float v8f;

// Only check the builtin on the DEVICE pass: hipcc compiles this TU for host
// too, where amdgcn builtins are (correctly) absent.
#if defined(__HIP_DEVICE_COMPILE__)
#if !__has_builtin(__builtin_amdgcn_wmma_f32_16x16x4_f32)
#error "wmma f32 16x16x4 builtin not available on gfx1250 device pass"
#endif
#endif

#define Hdim  1024
#define Wdim  1024
#define DMASK 1023
#define HWSZ  (1024 * 1024)

// db2 reconstruction filters (match reference constants exactly)
#define LO0 (-0.12940952255092145f)
#define LO1 ( 0.22414386804185735f)
#define LO2 ( 0.836516303737469f)
#define LO3 ( 0.48296291314469025f)
#define HI0 (-0.48296291314469025f)
#define HI1 ( 0.836516303737469f)
#define HI2 (-0.22414386804185735f)
#define HI3 (-0.12940952255092145f)

__device__ __forceinline__ float sel4(int idx, float f0, float f1, float f2, float f3) {
  float v = 0.0f;
  v = (idx == 0) ? f0 : v;
  v = (idx == 1) ? f1 : v;
  v = (idx == 2) ? f2 : v;
  v = (idx == 3) ? f3 : v;
  return v;
}

// ---------------------------------------------------------------------------
// Kernel 1: level-1 pass (s = 2).  Each thread handles one diagonal pixel
// (stencil) and one off-diagonal pixel (passthrough) in the same row.
// ---------------------------------------------------------------------------
__global__ __launch_bounds__(256) void swt_level1(const float* __restrict__ ll,
                                                  const float* __restrict__ lohi,
                                                  float* __restrict__ dst) {
  constexpr float FLO[4] = {LO0, LO1, LO2, LO3};
  constexpr float FHI[4] = {HI0, HI1, HI2, HI3};

  unsigned tid = blockIdx.x * 256u + threadIdx.x;   // 8 * 1024 * 512 threads
  int ch = tid & 511;            // diagonal column index / 2
  int r  = (tid >> 9) & DMASK;   // row
  int n  = tid >> 19;            // image

  const float* A  = ll   + (size_t)n * HWSZ;
  const float* Hb = lohi + ((size_t)n * 6 + 3) * HWSZ;  // level 1 -> channels 3,4,5
  const float* Vb = lohi + ((size_t)n * 6 + 4) * HWSZ;
  const float* Db = lohi + ((size_t)n * 6 + 5) * HWSZ;
  float* O = dst + (size_t)n * HWSZ;

  int i  = r & 1;
  int c  = 2 * ch + i;        // diagonal pixel (r%2 == c%2)
  int co = 2 * ch + 1 - i;    // off-diagonal pixel: passthrough

  O[r * Wdim + co] = A[r * Wdim + co];

  int tpar = ((r >> 1) ^ (c >> 1)) & 1;  // t_r xor t_c
  float acc = 0.0f;
#pragma unroll
  for (int u = 0; u < 4; ++u) {
    int rs = (r + 2 * u - 4 + Hdim) & DMASK;
    const float* Ar = A  + rs * Wdim;
    const float* Hr = Hb + rs * Wdim;
    const float* Vr = Vb + rs * Wdim;
    const float* Dr = Db + rs * Wdim;
    float loU = FLO[u], hiU = FHI[u];
    int v0 = (u ^ tpar) & 1;  // valid v parity
#pragma unroll
    for (int vv = 0; vv < 2; ++vv) {
      int cs = (c + 2 * v0 + 4 * vv - 4 + Wdim) & DMASK;
      float loV = v0 ? FLO[1 + 2 * vv] : FLO[2 * vv];
      float hiV = v0 ? FHI[1 + 2 * vv] : FHI[2 * vv];
      float av = Ar[cs], hv = Hr[cs], vvv = Vr[cs], dv = Dr[cs];
      acc += loV * (loU * av + hiU * hv) + hiV * (loU * vvv + hiU * dv);
    }
  }
  O[r * Wdim + c] = 0.5f * acc;
}

// ---------------------------------------------------------------------------
// Kernel 2: level-0 pass (s = 1), WMMA-based.
// Block = 128 threads (4 waves).  Block tile: 64 rows x 32 cols of output.
// LDS stages a (4 bands) x (64+4 rows) x (32+4 cols) halo region.
// Each wave owns a 16-row strip and produces two 16x16 output tiles:
//   column pass (VALU, 4 taps, full + alternated filters) produces the
//   A-matrices; row pass = 5 chunks of v_wmma_f32_16x16x4_f32 against
//   constant banded B matrices; epilogue combines full/alt with (-1)^{r+c}.
// ---------------------------------------------------------------------------
__global__ __launch_bounds__(128) void swt_level0_wmma(const float* __restrict__ interm,
                                                       const float* __restrict__ lohi,
                                                       float* __restrict__ out) {
  constexpr float FLO[4] = {LO0, LO1, LO2, LO3};
  constexpr float FHI[4] = {HI0, HI1, HI2, HI3};

  __shared__ float sm[4 * 68 * 36];  // [band][row 0..67][col 0..35], 39,168 B

  int b  = blockIdx.x;
  int xs = b & 31;          // 32 x-supertiles of width 32
  int yg = (b >> 5) & 7;    // 8 y-groups of 128 rows
  int n  = b >> 8;          // image
  int X0 = xs * 32;
  int Y0 = yg * 128;

  const float* base0 = interm + (size_t)n * HWSZ;             // cA  (level-1 result)
  const float* base1 = lohi + ((size_t)n * 6 + 0) * HWSZ;     // cH
  const float* base2 = lohi + ((size_t)n * 6 + 1) * HWSZ;     // cV
  const float* base3 = lohi + ((size_t)n * 6 + 2) * HWSZ;     // cD

  // ---- cooperative staging (coalesced along columns, wrapped) ----
  for (int idx = threadIdx.x; idx < 4 * 68 * 36; idx += 128) {
    int cc = idx % 36;
    int t  = (idx / 36) % 68;
    int bb = idx / (36 * 68);
    const float* p = (bb == 0) ? base0 : (bb == 1) ? base1 : (bb == 2) ? base2 : base3;
    int row = (Y0 - 2 + t + Hdim) & DMASK;
    int col = (X0 - 2 + cc + Wdim) & DMASK;
    sm[idx] = p[row * Wdim + col];
  }
  __syncthreads();

  int lane = threadIdx.x & 31;
  int wv   = threadIdx.x >> 5;  // wave id 0..3 -> 16-row strip
  int l15  = lane & 15;
  int g    = lane >> 4;         // lane half: selects K = {0,1} vs {2,3}

  // ---- constant banded B matrices (4x16 f32, 2 VGPRs each) ----
  // Assumed layout (mirrors documented A/C/D striping): reg r, lane-half g
  //   -> K = r + 2g, N = lane&15.   B_k[K][N] = filt[(4k+K) - N] (0 outside).
  v2f bLO[5], bHI[5], bLOa[5], bHIa[5];
#pragma unroll
  for (int k = 0; k < 5; ++k) {
    float lo[2], hi[2], loa[2], hia[2];
#pragma unroll
    for (int rr = 0; rr < 2; ++rr) {
      int e   = 4 * k + rr + 2 * g;
      int idx = e - l15;
      float l = sel4(idx, LO0, LO1, LO2, LO3);
      float h = sel4(idx, HI0, HI1, HI2, HI3);
      float s = (idx == 1 || idx == 3) ? -1.0f : 1.0f;  // alt = (-1)^idx
      lo[rr] = l;      hi[rr] = h;
      loa[rr] = s * l; hia[rr] = s * h;
    }
    bLO[k]  = (v2f){lo[0], lo[1]};
    bHI[k]  = (v2f){hi[0], hi[1]};
    bLOa[k] = (v2f){loa[0], loa[1]};
    bHIa[k] = (v2f){hia[0], hia[1]};
  }

  float* outp = out + (size_t)n * HWSZ;

#pragma unroll
  for (int tau = 0; tau < 2; ++tau) {  // two 16-wide tiles per wave
    v8f accF = {0.f, 0.f, 0.f, 0.f, 0.f, 0.f, 0.f, 0.f};
    v8f accA = {0.f, 0.f, 0.f, 0.f, 0.f, 0.f, 0.f, 0.f};

#pragma unroll
    for (int k = 0; k < 5; ++k) {
      // Column pass: each lane computes exactly the A-matrix elements it
      // owns in the documented 16x4 f32 A layout:
      //   lane<16: M=lane, K=0,1 ; lane>=16: M=lane-16, K=2,3.
      float U1[2], U2[2], U1a[2], U2a[2];
#pragma unroll
      for (int rr = 0; rr < 2; ++rr) {
        int cc = 16 * tau + 4 * k + rr + 2 * g;  // <= 35
        float u1 = 0.f, u2 = 0.f, u1a = 0.f, u2a = 0.f;
#pragma unroll
        for (int u = 0; u < 4; ++u) {
          int t = wv * 16 + l15 + u;  // LDS row (halo-offset already folded)
          float av = sm[(0 * 68 + t) * 36 + cc];
          float hv = sm[(1 * 68 + t) * 36 + cc];
          float vv = sm[(2 * 68 + t) * 36 + cc];
          float dv = sm[(3 * 68 + t) * 36 + cc];
          float t1 = FLO[u] * av + FHI[u] * hv;  // LO-col(A) + HI-col(H)
          float t2 = FLO[u] * vv + FHI[u] * dv;  // LO-col(V) + HI-col(D)
          u1 += t1;
          u2 += t2;
          if (u & 1) { u1a -= t1; u2a -= t2; } else { u1a += t1; u2a += t2; }
        }
        U1[rr] = u1; U2[rr] = u2; U1a[rr] = u1a; U2a[rr] = u2a;
      }
      v2f aF1 = (v2f){U1[0],  U1[1]};
      v2f aF2 = (v2f){U2[0],  U2[1]};
      v2f aA1 = (v2f){U1a[0], U1a[1]};
      v2f aA2 = (v2f){U2a[0], U2a[1]};

      // Row pass: D = A x B + C, f32 exact.
      accF = __builtin_amdgcn_wmma_f32_16x16x4_f32(false, aF1, false, bLO[k],
                                                   (short)0, accF, false, false);
      accF = __builtin_amdgcn_wmma_f32_16x16x4_f32(false, aF2, false, bHI[k],
                                                   (short)0, accF, false, false);
      accA = __builtin_amdgcn_wmma_f32_16x16x4_f32(false, aA1, false, bLOa[k],
                                                   (short)0, accA, false, false);
      accA = __builtin_amdgcn_wmma_f32_16x16x4_f32(false, aA2, false, bHIa[k],
                                                   (short)0, accA, false, false);
    }

    // Epilogue: out = 0.25*(S_full + (-1)^{r+c} S_alt), per documented C/D
    // layout: VGPR j holds row M=j (lanes 0-15) / M=j+8 (lanes 16-31).
    int x_out = X0 + 16 * tau + l15;
#pragma unroll
    for (int j = 0; j < 8; ++j) {
      int y_out  = Y0 + wv * 16 + j + 8 * g;
      float sg   = ((y_out + x_out) & 1) ? -0.25f : 0.25f;
      outp[y_out * Wdim + x_out] = 0.25f * accF[j] + sg * accA[j];
    }
  }
}

// ---------------------------------------------------------------------------
extern "C" void kernel_launch(void* const* d_in, const int* in_sizes, int n_in,
                              void* d_out, int out_size, void* d_ws, size_t ws_size,
                              hipStream_t stream) {
  const float* ll   = (const float*)d_in[0];  // (8,1,1024,1024)
  const float* lohi = (const float*)d_in[1];  // (8,6,1024,1024)
  float* interm = (float*)d_ws;               // 32 MB intermediate
  float* out    = (float*)d_out;

  // Level-1 pass: 8 * 1024 * 512 threads, each: 1 stencil px + 1 copy px.
  swt_level1<<<16384, 256, 0, stream>>>(ll, lohi, interm);

  // Level-0 pass: 32 x-supertiles * 8 y-groups * 16 values of n (b >> 8) = 4096 blocks.
  swt_level0_wmma<<<4096, 128, 0, stream>>>(interm, lohi, out);
}